// Quantiser_65249143161578
// MI455X (gfx1250) — compile-verified
//
#include <hip/hip_runtime.h>

typedef __attribute__((ext_vector_type(2))) float v2f;
typedef __attribute__((ext_vector_type(8))) float v8f;

#define NUM_EMB 1024
#define EMB_DIM 64
#define N_ROWS  (32 * 64 * 64)   // 131072 spatial positions
#define HW      4096             // 64*64, channel stride in NCHW
#define TILES_PER_BLOCK 8        // one 16-row tile per wave, 8 waves/block

// ---------------------------------------------------------------------------
// Prep: zero loss accumulator, compute 0.5*||e_k||^2 for each code.
// ---------------------------------------------------------------------------
__global__ __launch_bounds__(256) void vq_prep(const float* __restrict__ emb,
                                               float* __restrict__ halfnorm,
                                               float* __restrict__ out) {
  int k = blockIdx.x * 256 + threadIdx.x;
  if (k == 0) out[0] = 0.0f;               // loss slot (harness poisons d_out)
  if (k < NUM_EMB) {
    const float* e = emb + k * EMB_DIM;
    float s = 0.0f;
#pragma unroll 8
    for (int c = 0; c < EMB_DIM; ++c) { float v = e[c]; s += v * v; }
    halfnorm[k] = 0.5f * s;
  }
}

// ---------------------------------------------------------------------------
// Main: nearest-code search via V_WMMA_F32_16X16X4_F32, gather + loss.
// argmin_k ||x-e_k||^2  ==  argmax_k (x.e_k - 0.5*||e_k||^2)
// Two code tiles per iteration -> two independent WMMA accumulation chains
// to keep the f32 XDL pipe busy at low wave occupancy (LDS-limited).
// ---------------------------------------------------------------------------
__global__ __launch_bounds__(256) void vq_main(const float* __restrict__ in,
                                               const float* __restrict__ emb,
                                               const float* __restrict__ halfnorm,
                                               float* __restrict__ out) {
  extern __shared__ float smem[];                 // [NUM_EMB*EMB_DIM] + idx scratch
  float* semb = smem;
  int*   sidx = (int*)(smem + NUM_EMB * EMB_DIM); // 8 waves * 16 rows

  const int tid = threadIdx.x;

  // Stage full codebook (256 KB) into LDS: B128 loads + DS stores.
  {
    const float4* src = (const float4*)emb;
    float4*       dst = (float4*)semb;
#pragma unroll 4
    for (int i = tid; i < NUM_EMB * EMB_DIM / 4; i += 256) dst[i] = src[i];
  }
  __syncthreads();

  const int lane = tid & 31;
  const int wv   = tid >> 5;
  const int tile = blockIdx.x * TILES_PER_BLOCK + wv;   // 16-row tile
  const int row0 = tile * 16;
  const int bb   = row0 >> 12;                          // batch index
  const int nm   = row0 & 4095;                         // h*64+w base
  const float* inb = in + (size_t)bb * (EMB_DIM * HW) + nm;

  const int half = lane >> 4;   // 0: K pair {0,1}, 1: K pair {2,3} (A layout)
  const int l15  = lane & 15;   // A: row M ; B/C/D: column N
  const int coff = half * 2;

  // A tile in registers: 16 chunks of 4 channels. ISA 16x4 f32 A layout:
  // lanes 0-15 hold (M=lane, K=0/1), lanes 16-31 hold (M=lane-16, K=2/3).
  v2f a[16];
#pragma unroll
  for (int kk = 0; kk < 16; ++kk) {
    const int c = kk * 4 + coff;
    v2f t;
    t.x = inb[(size_t)c * HW + l15];
    t.y = inb[(size_t)(c + 1) * HW + l15];
    a[kk] = t;
  }

  float bestv[8];
  int   besti[8];
#pragma unroll
  for (int r = 0; r < 8; ++r) { bestv[r] = -3.4e38f; besti[r] = 0; }

  // Sweep 32 iterations x 2 code tiles; 2x16 interleaved f32 WMMA chains.
  for (int ct = 0; ct < 32; ++ct) {
    const int cb0 = ct * 32;          // first tile: codes cb0..cb0+15
    const int cb1 = cb0 + 16;         // second tile: codes cb1..cb1+15
    v8f acc0 = {};
    v8f acc1 = {};
    // B layout mirrors A: lane = column N (code), half selects K pair.
    const float* bb0 = semb + (size_t)(cb0 + l15) * EMB_DIM + coff;
    const float* bb1 = semb + (size_t)(cb1 + l15) * EMB_DIM + coff;
#pragma unroll
    for (int kk = 0; kk < 16; ++kk) {
      const v2f bv0 = *(const v2f*)(bb0 + kk * 4);   // ds_load_b64 from LDS
      const v2f bv1 = *(const v2f*)(bb1 + kk * 4);
      acc0 = __builtin_amdgcn_wmma_f32_16x16x4_f32(
          false, a[kk], false, bv0, (short)0, acc0, false, false);
      acc1 = __builtin_amdgcn_wmma_f32_16x16x4_f32(
          false, a[kk], false, bv1, (short)0, acc1, false, false);
    }
    const float hn0 = halfnorm[cb0 + l15];
    const float hn1 = halfnorm[cb1 + l15];
    const int   c0  = cb0 + l15;
    const int   c1  = cb1 + l15;
#pragma unroll
    for (int r = 0; r < 8; ++r) {                 // tile0 first: lower codes
      const float s0 = acc0[r] - hn0;             // win ties (strict >)
      if (s0 > bestv[r]) { bestv[r] = s0; besti[r] = c0; }
      const float s1 = acc1[r] - hn1;
      if (s1 > bestv[r]) { bestv[r] = s1; besti[r] = c1; }
    }
  }

  // Reduce argmax across the 16 lanes sharing each row (N dimension).
  // Tie-break to lower code index to match jnp.argmin first-occurrence.
#pragma unroll
  for (int r = 0; r < 8; ++r) {
#pragma unroll
    for (int m = 1; m <= 8; m <<= 1) {
      const float ov = __shfl_xor(bestv[r], m, 32);
      const int   oi = __shfl_xor(besti[r], m, 32);
      if (ov > bestv[r] || (ov == bestv[r] && oi < besti[r])) {
        bestv[r] = ov; besti[r] = oi;
      }
    }
    // C/D layout: VGPR r, lanes 0-15 -> row r, lanes 16-31 -> row 8+r.
    if (l15 == r) sidx[wv * 16 + half * 8 + r] = besti[r];
  }
  __syncthreads();

  // Write pass: lane handles (row = l15, channels half*32..half*32+31).
  const int   rr    = l15;
  const int   myidx = sidx[wv * 16 + rr];
  const float* qrow = semb + (size_t)myidx * EMB_DIM;
  float* outb = out + 1 + (size_t)bb * (EMB_DIM * HW) + nm;  // q_st, NCHW
  float d2 = 0.0f;
#pragma unroll 4
  for (int j = 0; j < 32; ++j) {
    const int c = half * 32 + j;
    const float q = qrow[c];                       // gather winning code
    const float x = inb[(size_t)c * HW + rr];
    outb[(size_t)c * HW + rr] = q;                 // straight-through fwd
    const float d = q - x;
    d2 += d * d;
  }

  // loss = (1 + commitment) * mean((q-x)^2); wave-reduce then atomic add.
#pragma unroll
  for (int m = 1; m < 32; m <<= 1) d2 += __shfl_xor(d2, m, 32);
  if (lane == 0)
    atomicAdd(out, d2 * (1.25f / (float)(N_ROWS * EMB_DIM)));
}

// ---------------------------------------------------------------------------
extern "C" void kernel_launch(void* const* d_in, const int* in_sizes, int n_in,
                              void* d_out, int out_size, void* d_ws, size_t ws_size,
                              hipStream_t stream) {
  const float* in  = (const float*)d_in[0];   // (32,64,64,64) f32
  const float* emb = (const float*)d_in[1];   // (1024,64) f32
  float* out = (float*)d_out;                 // [0]=loss, [1..]=q_st NCHW
  float* hn  = (float*)d_ws;                  // 1024 floats

  vq_prep<<<dim3((NUM_EMB + 255) / 256), dim3(256), 0, stream>>>(emb, hn, out);

  const size_t shmem = (size_t)NUM_EMB * EMB_DIM * sizeof(float) +
                       TILES_PER_BLOCK * 16 * sizeof(int);   // ~256.5 KB (<320 KB WGP LDS)
  const int nblocks = N_ROWS / 16 / TILES_PER_BLOCK;         // 1024
  vq_main<<<dim3(nblocks), dim3(256), shmem, stream>>>(in, emb, hn, out);
}